// WINDOW_WARPING_16698832847536
// MI455X (gfx1250) — compile-verified
//
#include <hip/hip_runtime.h>
#include <stdint.h>

// Problem geometry (from reference)
#define BATCH   256
#define CHAN    64
#define SLEN    8192
#define WSTART  4500
#define NWIN    750            // END - START
#define MINT    746            // reduced tridiagonal system size (NWIN - 4)
#define NROWS   (BATCH*CHAN)   // 16384 independent spline rows
#define MPITCH  752            // workspace row pitch in floats (16B aligned)

// float4 tiling of the S dimension
#define F4_PER_ROW    (SLEN/4)     // 2048
#define WIN_F4_FIRST  (WSTART/4)   // 1125 (4500 % 4 == 0 -> f4 aligned window)
#define WIN_F4_COUNT  188          // f4 cols 1125..1312 cover elements 4500..5251
#define COPY_F4_PER_ROW (F4_PER_ROW - WIN_F4_COUNT)  // 1860

// WMMA banded-inverse solve parameters
#define MROWSPG 16                 // spline rows per workgroup (one wave)
#define HBAND   32                 // half bandwidth (rho^33 ~ 1.3e-19, exact in f32)
#define NTILES  47                 // ceil(746/16) output column tiles
#define NCHUNK  20                 // 80 j-values per tile / K=4
#define DPITCH  822                // LDS d-row pitch (even for b64 align, 16-row bank-clean)
#define JPAD    32                 // left zero pad of d rows
#define JMAX    784                // stage j in [-32, 784): covers max q = 783
#define LNRHO     (-1.3169578969248166f)  // log(2 - sqrt(3))
#define INV2SQRT3 (0.28867513459481287f)  // 1/(2*sqrt(3))

typedef float v2f __attribute__((ext_vector_type(2)));
typedef float v8f __attribute__((ext_vector_type(8)));

// ---------------------------------------------------------------------------
// Kernel 1: spline second-derivative solve via WMMA banded inverse.
//   Not-a-knot elimination (exact):  M1 = r[1]/6,  M748 = r[748]/6.
//   Interior M[2..747] = T^-1 d,  T = tridiag(1,4,1)_746,
//     T^-1[p,q] = (-1)^{p+q} rho^{|p-q|}(1-rho^{2(min+1)})(1-rho^{2(m-max)})/(2*sqrt(3)).
//   Banded to |p-q|<=32 (exact at f32). Per 16-row group: 47 column tiles,
//   each = 20 x V_WMMA_F32_16X16X4_F32 accumulations.
//   Interior tiles (3..42): branch-free fully-unrolled loop, Toeplitz weights
//   from LDS. Boundary tiles (0,1,2,43..46): exact corrected weights per lane.
// ---------------------------------------------------------------------------
__global__ __launch_bounds__(32)
void WW_wmma_solve_kernel(const float* __restrict__ x, float* __restrict__ Mws) {
    __shared__ float dls[MROWSPG * DPITCH];   // 52,608 B  d rows, zero padded
    __shared__ float Btab[NCHUNK * 64];       //  5,120 B  [chunk][n][k] Toeplitz weights
    __shared__ float Mb0[16 * 16];            //  1,024 B  tile 0 stash (for M2)
    __shared__ float MbL[16 * 16];            //  1,024 B  tile 46 stash (for M747)
    __shared__ float M1s[16], M748s[16];

    const int lane  = threadIdx.x;     // 0..31, one wave, EXEC all-ones at WMMAs
    const int nl    = lane & 15;
    const int hh    = lane >> 4;       // half-wave: selects K pair {2h, 2h+1}
    const int rbase = blockIdx.x * MROWSPG;

    // boundary second derivatives per row (exact): M1 = r[1]/6 = stencil
    if (lane < 16) {
        const float* w = x + (size_t)(rbase + lane) * SLEN + WSTART;
        M1s[lane]   = w[2]   - 2.0f * w[1]   + w[0];
        M748s[lane] = w[749] - 2.0f * w[748] + w[747];
    }
    __syncthreads();

    // Toeplitz weight table: Btab[c][n][k] = wT(4c - 32 + k - n)
    for (int c = 0; c < NCHUNK; ++c) {
        #pragma unroll
        for (int kk = 0; kk < 2; ++kk) {
            const int k     = 2 * hh + kk;
            const int delta = 4 * c - HBAND + k - nl;
            const int ad    = delta < 0 ? -delta : delta;
            float wgt = __expf(LNRHO * (float)ad) * INV2SQRT3;
            if (delta & 1) wgt = -wgt;
            Btab[c * 64 + nl * 4 + k] = wgt;
        }
    }

    // stage d (RHS) rows into LDS, zero padded: dls[r][j+32], j in [-32, 784)
    for (int r = 0; r < MROWSPG; ++r) {
        const float* w = x + (size_t)(rbase + r) * SLEN + WSTART;
        for (int j = lane - JPAD; j < JMAX; j += 32) {
            float dv = 0.0f;
            if (j >= 0 && j < MINT) {
                dv = 6.0f * (w[j + 3] - 2.0f * w[j + 2] + w[j + 1]);
                if (j == 0)        dv -= M1s[r];
                if (j == MINT - 1) dv -= M748s[r];
            }
            dls[r * DPITCH + (j + JPAD)] = dv;
        }
    }
    __syncthreads();

    // ---- interior tiles: branch-free, fully unrolled (20 WMMA per tile) ----
    for (int t = 3; t <= 42; ++t) {
        const int i0 = t * 16;
        v8f acc = {0.f, 0.f, 0.f, 0.f, 0.f, 0.f, 0.f, 0.f};
        #pragma unroll
        for (int c = 0; c < NCHUNK; ++c) {
            const int j0 = i0 - HBAND + 4 * c;
            const v2f a = *(const v2f*)&dls[nl * DPITCH + (j0 + 2 * hh + JPAD)];
            const v2f b = *(const v2f*)&Btab[c * 64 + nl * 4 + 2 * hh];
            acc = __builtin_amdgcn_wmma_f32_16x16x4_f32(
                      false, a, false, b, (short)0, acc, false, false);
        }
        // cols i0..i0+15 <= 687 < MINT: unconditional, coalesced per-row stores
        float* mp = Mws + (size_t)(rbase + 8 * hh) * MPITCH + 2 + i0 + nl;
        #pragma unroll
        for (int r = 0; r < 8; ++r)
            mp[(size_t)r * MPITCH] = acc[r];
    }

    // ---- boundary tiles: exact corrected weights (corrections decay as rho^2k)
    const int btiles[7] = {0, 1, 2, 43, 44, 45, 46};
    #pragma unroll
    for (int bt = 0; bt < 7; ++bt) {
        const int t  = btiles[bt];
        const int i0 = t * 16;
        v8f acc = {0.f, 0.f, 0.f, 0.f, 0.f, 0.f, 0.f, 0.f};
        for (int c = 0; c < NCHUNK; ++c) {
            const int j0 = i0 - HBAND + 4 * c;
            const v2f a = *(const v2f*)&dls[nl * DPITCH + (j0 + 2 * hh + JPAD)];
            v2f b;
            const int p = i0 + nl;    // output position (column of this tile)
            #pragma unroll
            for (int kk = 0; kk < 2; ++kk) {
                const int q = j0 + 2 * hh + kk;
                float wgt = 0.0f;
                if (q >= 0 && q < MINT) {
                    const int mn = p < q ? p : q;
                    const int mx = p < q ? q : p;
                    const int ad = p < q ? q - p : p - q;
                    float v = __expf(LNRHO * (float)ad);
                    v *= 1.0f - __expf(LNRHO * (float)(2 * (mn + 1)));
                    v *= 1.0f - __expf(LNRHO * (float)(2 * (MINT - mx)));
                    v *= INV2SQRT3;
                    wgt = ((p + q) & 1) ? -v : v;
                }
                b[kk] = wgt;
            }
            acc = __builtin_amdgcn_wmma_f32_16x16x4_f32(
                      false, a, false, b, (short)0, acc, false, false);
        }
        const int col = i0 + nl;
        if (col < MINT) {
            float* mp = Mws + (size_t)(rbase + 8 * hh) * MPITCH + 2 + col;
            #pragma unroll
            for (int r = 0; r < 8; ++r)
                mp[(size_t)r * MPITCH] = acc[r];
        }
        if (t == 0) {
            #pragma unroll
            for (int r = 0; r < 8; ++r) Mb0[(r + 8 * hh) * 16 + nl] = acc[r];
        }
        if (t == NTILES - 1) {
            #pragma unroll
            for (int r = 0; r < 8; ++r) MbL[(r + 8 * hh) * 16 + nl] = acc[r];
        }
    }
    __syncthreads();

    // not-a-knot closure: M0 = 2*M1 - M2 ; M749 = 2*M748 - M747
    if (lane < 16) {
        const float M1   = M1s[lane];
        const float M748 = M748s[lane];
        const float M2   = Mb0[lane * 16 + 0];   // tile 0,  col 0  -> spline idx 2
        const float M747 = MbL[lane * 16 + 9];   // tile 46, col 9  -> spline idx 747
        float* mp = Mws + (size_t)(rbase + lane) * MPITCH;
        mp[0]   = 2.0f * M1 - M2;
        mp[1]   = M1;
        mp[748] = M748;
        mp[749] = 2.0f * M748 - M747;
    }
}

// ---------------------------------------------------------------------------
// Kernel 2: bulk copy of the non-window region (488 MB each way) through the
// CDNA5 async LDS path: global -> LDS (ASYNCcnt) -> global, no VGPR data.
// ---------------------------------------------------------------------------
__global__ __launch_bounds__(256)
void WW_copy_async_kernel(const float* __restrict__ x, float* __restrict__ out) {
    __shared__ float4 stage[256];
    const int tid = threadIdx.x;
    const int c   = blockIdx.x * 256 + tid;        // 0..2047 (8 blocks in x)
    if (c >= COPY_F4_PER_ROW) return;              // 1860 active copy columns
    const int f4col = (c < WIN_F4_FIRST) ? c : c + WIN_F4_COUNT;  // skip window
    const size_t f4idx = (size_t)blockIdx.y * F4_PER_ROW + f4col;

    const unsigned long long srcA = (unsigned long long)(uintptr_t)((const float4*)x + f4idx);
    const unsigned long long dstA = (unsigned long long)(uintptr_t)((float4*)out + f4idx);
    // low 32 bits of a generic LDS pointer == workgroup-relative LDS offset
    const unsigned ldsOff = (unsigned)(uintptr_t)(&stage[tid]);

    asm volatile(
        "global_load_async_to_lds_b128 %0, %1, off\n\t"
        "s_wait_asynccnt 0x0\n\t"
        "global_store_async_from_lds_b128 %2, %0, off\n\t"
        "s_wait_asynccnt 0x0"
        :
        : "v"(ldsOff), "v"(srcA), "v"(dstA)
        : "memory");
}

// ---------------------------------------------------------------------------
// Kernel 3: window evaluation (one thread per output float4, coalesced b128
// store; gathers hit the L2-resident window / M workspace).
// ---------------------------------------------------------------------------
__global__ __launch_bounds__(192)
void WW_eval_kernel(const float* __restrict__ x,
                    const float* __restrict__ rand_apply,
                    const float* __restrict__ scale_rand,
                    const float* __restrict__ Mws,
                    float* __restrict__ out) {
    const int q   = threadIdx.x;          // 0..187 active
    const int row = blockIdx.x;           // 0..16383
    if (q >= WIN_F4_COUNT) return;
    const int b = row >> 6;               // row / CHAN

    const float* __restrict__ w  = x   + (size_t)row * SLEN + WSTART;
    const float* __restrict__ Mr = Mws + (size_t)row * MPITCH;

    const float4 xv   = *(const float4*)(w + 4 * q);
    const bool  apply = rand_apply[b] < 0.5f;
    const float scale = 0.7f + 0.6f * scale_rand[b];   // 1 - ws + 2*ws*sr, ws=0.3

    float xc[4] = {xv.x, xv.y, xv.z, xv.w};
    float rc[4];
    #pragma unroll
    for (int e = 0; e < 4; ++e) {
        const int i = 4 * q + e;
        if (i >= NWIN || !apply) {
            rc[e] = xc[e];                 // pass-through
        } else {
            float t = fminf((float)i * scale, (float)(NWIN - 1));
            int idx = (int)t;              // t >= 0, trunc == floor
            if (idx > NWIN - 2) idx = NWIN - 2;
            const float u  = t - (float)idx;
            const float y0 = w[idx],  y1 = w[idx + 1];
            const float m0 = Mr[idx], m1 = Mr[idx + 1];
            const float bc = (y1 - y0) - (2.0f * m0 + m1) * (1.0f / 6.0f);
            rc[e] = y0 + u * (bc + u * (0.5f * m0 + u * (m1 - m0) * (1.0f / 6.0f)));
        }
    }
    float4 ov; ov.x = rc[0]; ov.y = rc[1]; ov.z = rc[2]; ov.w = rc[3];
    *(float4*)(out + (size_t)row * SLEN + WSTART + 4 * q) = ov;
}

// ---------------------------------------------------------------------------
// Host launcher. Workspace: NROWS * MPITCH * 4 = 49,283,072 B (~47 MB).
// ---------------------------------------------------------------------------
extern "C" void kernel_launch(void* const* d_in, const int* in_sizes, int n_in,
                              void* d_out, int out_size, void* d_ws, size_t ws_size,
                              hipStream_t stream) {
    const float* x          = (const float*)d_in[0];
    const float* rand_apply = (const float*)d_in[1];
    const float* scale_rand = (const float*)d_in[2];
    float*       out        = (float*)d_out;
    float*       Mws        = (float*)d_ws;

    // 1) WMMA banded-inverse spline solve (writes Mws)
    WW_wmma_solve_kernel<<<NROWS / MROWSPG, 32, 0, stream>>>(x, Mws);

    // 2) bulk copy of everything outside the window (async LDS staging)
    dim3 cg((F4_PER_ROW + 255) / 256, NROWS);      // (8, 16384)
    WW_copy_async_kernel<<<cg, 256, 0, stream>>>(x, out);

    // 3) window warp evaluation (reads Mws, writes window region of out)
    WW_eval_kernel<<<NROWS, 192, 0, stream>>>(x, rand_apply, scale_rand, Mws, out);
}